// HNetLoss_80083960201790
// MI455X (gfx1250) — compile-verified
//
#include <hip/hip_runtime.h>

typedef __attribute__((ext_vector_type(2))) float v2f;
typedef __attribute__((ext_vector_type(8))) float v8f;

#define HNET_N 512
#define PPL    16   // points per lane: N / 32 (wave32)

// One wave (32 lanes) per batch. Single pass over gt_pts; moments reduced
// across lanes with V_WMMA_F32_16X16X4_F32; 4x4 solve in f64 per-lane.
__global__ __launch_bounds__(32) void hnet_batch_kernel(
    const float* __restrict__ gt,   // (B, N, 3) f32
    const float* __restrict__ tc,   // (B, 6)    f32
    float* __restrict__ bsum)       // (B,) per-batch sum of squared residuals
{
  const int b    = blockIdx.x;
  const int lane = threadIdx.x & 31;

  const float* P = gt + (size_t)b * (HNET_N * 3);
  const float* C = tc + (size_t)b * 6;

  const float c0 = C[0], c1 = C[1], c2 = C[2];
  const float c3 = C[3], c4 = C[4], c5 = C[5];

  float xs[PPL], Yv[PPL], zs[PPL];
  float S1 = 0.f, S2 = 0.f, S3 = 0.f, S4 = 0.f, S5 = 0.f, S6 = 0.f;
  float T0 = 0.f, T1 = 0.f, T2 = 0.f, T3 = 0.f;

#pragma unroll
  for (int i = 0; i < PPL; ++i) {
    const int n   = lane + 32 * i;            // coalesced b96 loads
    const float x = P[3 * n + 0];
    const float y = P[3 * n + 1];
    const float z = P[3 * n + 2];
    const float tp0 = c0 * x + c1 * y + c2 * z;
    const float tp1 = c3 * y + c4 * z;
    const float tp2 = c5 * y + z;
    const float X = tp0 / tp2;
    const float Y = tp1 / tp2;
    xs[i] = x; Yv[i] = Y; zs[i] = z;
    const float Y2 = Y * Y, Y3 = Y2 * Y;
    S1 += Y;       S2 += Y2;      S3 += Y3;
    S4 += Y2 * Y2; S5 += Y2 * Y3; S6 += Y3 * Y3;
    T0 += X; T1 += X * Y; T2 += X * Y2; T3 += X * Y3;
  }

  // ---- cross-lane sum of the 10 moments via V_WMMA_F32_16X16X4_F32 ----
  // part[lane][0..9] = {S1..S6, T0..T3}, cols 10..15 zero-padded.
  __shared__ float sm[32][16];
  sm[lane][0] = S1; sm[lane][1] = S2; sm[lane][2] = S3;
  sm[lane][3] = S4; sm[lane][4] = S5; sm[lane][5] = S6;
  sm[lane][6] = T0; sm[lane][7] = T1; sm[lane][8] = T2; sm[lane][9] = T3;
#pragma unroll
  for (int j = 10; j < 16; ++j) sm[lane][j] = 0.f;
  __syncthreads();

  // D = sum_t Ones(16x4) * B_t(4x16):  D[m][n] = sum over all 32 lanes of part[lane][n]
  // B layout (4x16, K x N): VGPR r, lane l holds B(k = 2*(l>=16)+r, n = l&15).
  const int f  = lane & 15;
  const int hi = lane >> 4;
  v2f aones; aones[0] = 1.0f; aones[1] = 1.0f;
  v8f acc = {0.f, 0.f, 0.f, 0.f, 0.f, 0.f, 0.f, 0.f};
#pragma unroll
  for (int t = 0; t < 8; ++t) {
    v2f bf;
    bf[0] = sm[4 * t + 2 * hi + 0][f];
    bf[1] = sm[4 * t + 2 * hi + 1][f];
    acc = __builtin_amdgcn_wmma_f32_16x16x4_f32(false, aones, false, bf,
                                                (short)0, acc, false, false);
  }
  // Row M=0 lives in VGPR0, lanes 0..15 hold N=0..15 -> lane j has moment j.
  const float red0 = acc[0];

  double S[7];
  S[0] = (double)HNET_N;
  S[1] = (double)__shfl(red0, 0, 32);
  S[2] = (double)__shfl(red0, 1, 32);
  S[3] = (double)__shfl(red0, 2, 32);
  S[4] = (double)__shfl(red0, 3, 32);
  S[5] = (double)__shfl(red0, 4, 32);
  S[6] = (double)__shfl(red0, 5, 32);
  const double R0 = (double)__shfl(red0, 6, 32);
  const double R1 = (double)__shfl(red0, 7, 32);
  const double R2 = (double)__shfl(red0, 8, 32);
  const double R3 = (double)__shfl(red0, 9, 32);

  // G[k][l] = S[6-k-l]; rhs[k] = T_{3-k}. Solve G w = rhs (SPD, f64, uniform).
  double Mx[4][5];
#pragma unroll
  for (int k = 0; k < 4; ++k)
#pragma unroll
    for (int l = 0; l < 4; ++l) Mx[k][l] = S[6 - k - l];
  Mx[0][4] = R3; Mx[1][4] = R2; Mx[2][4] = R1; Mx[3][4] = R0;

#pragma unroll
  for (int k = 0; k < 4; ++k) {
    const double ip = 1.0 / Mx[k][k];
#pragma unroll
    for (int j = k; j < 5; ++j) Mx[k][j] *= ip;
#pragma unroll
    for (int i2 = k + 1; i2 < 4; ++i2) {
      const double fac = Mx[i2][k];
#pragma unroll
      for (int j = k; j < 5; ++j) Mx[i2][j] -= fac * Mx[k][j];
    }
  }
  const double w3d = Mx[3][4];
  const double w2d = Mx[2][4] - Mx[2][3] * w3d;
  const double w1d = Mx[1][4] - Mx[1][2] * w2d - Mx[1][3] * w3d;
  const double w0d = Mx[0][4] - Mx[0][1] * w1d - Mx[0][2] * w2d - Mx[0][3] * w3d;
  const float w0 = (float)w0d, w1 = (float)w1d, w2 = (float)w2d, w3 = (float)w3d;

  // Row 0 of H^{-1}: H = [[a,b,c],[0,d,e],[0,f,1]], det = a*(d - e*f)
  const float dm   = c3 - c4 * c5;
  const float idet = 1.0f / (c0 * dm);
  const float i00  = dm * idet;
  const float i01  = -(c1 - c2 * c5) * idet;
  const float i02  = (c1 * c4 - c2 * c3) * idet;

  float racc = 0.f;
#pragma unroll
  for (int i = 0; i < PPL; ++i) {
    const float Y  = Yv[i];
    const float xp = ((w0 * Y + w1) * Y + w2) * Y + w3;
    const float lb = zs[i] * (i00 * xp + i01 * Y + i02);
    const float d  = xs[i] - lb;
    racc += d * d;
  }
#pragma unroll
  for (int off = 16; off > 0; off >>= 1)
    racc += __shfl_xor(racc, off, 32);
  if (lane == 0) bsum[b] = racc;
}

// Deterministic fixed-order final reduction: sum(bsum)/ (B*N)
__global__ __launch_bounds__(256) void hnet_reduce_kernel(
    const float* __restrict__ bsum, int B, float* __restrict__ out)
{
  __shared__ double sred[256];
  const int t = threadIdx.x;
  double s = 0.0;
  for (int k = t; k < B; k += 256) s += (double)bsum[k];
  sred[t] = s;
  __syncthreads();
  for (int stride = 128; stride > 0; stride >>= 1) {
    if (t < stride) sred[t] += sred[t + stride];
    __syncthreads();
  }
  if (t == 0) out[0] = (float)(sred[0] / ((double)B * (double)HNET_N));
}

extern "C" void kernel_launch(void* const* d_in, const int* in_sizes, int n_in,
                              void* d_out, int out_size, void* d_ws, size_t ws_size,
                              hipStream_t stream) {
  (void)n_in; (void)out_size; (void)ws_size;
  const float* gt = (const float*)d_in[0];   // (B, N, 3)
  const float* tc = (const float*)d_in[1];   // (B, 6)
  const int B = in_sizes[1] / 6;             // 16384
  float* bsum = (float*)d_ws;                // B floats of scratch
  float* out  = (float*)d_out;

  hipLaunchKernelGGL(hnet_batch_kernel, dim3(B), dim3(32), 0, stream, gt, tc, bsum);
  hipLaunchKernelGGL(hnet_reduce_kernel, dim3(1), dim3(256), 0, stream, bsum, B, out);
}